// TverskyProjection_52097953300603
// MI455X (gfx1250) — compile-verified
//
#include <hip/hip_runtime.h>

// ---------------------------------------------------------------------------
// TverskyProjection for MI455X (gfx1250, wave32, WMMA).
//
// Decomposition: with p=relu(a), ha=1[a>0], q=relu(proto), hb=1[proto>0]:
//   inter = sum_f min(p,q)                       (elementwise, VALU)
//   S_a   = sum_f p*hb   -> f32 WMMA GEMM
//   S_b   = sum_f ha*q   -> f32 WMMA GEMM
//   out   = (theta+alpha+beta)*inter - alpha*S_a - beta*S_b
// All matrix math uses V_WMMA_F32_16X16X4_F32 (exact f32, matches reference).
// ---------------------------------------------------------------------------

typedef float v2f  __attribute__((ext_vector_type(2)));
typedef float v4f  __attribute__((ext_vector_type(4)));
typedef float v8f  __attribute__((ext_vector_type(8)));

#define ROWS  2048   // B*S
#define INF   1024
#define NF    128
#define OUTF  512

// ---------------------------------------------------------------------------
// K1: a = x @ fb  (M=2048, K=1024, N=128), fused relu epilogue -> p, ha.
// One wave per 16x16 output tile. A-fragment layout per ISA 16x4 f32 A-matrix:
//   lane L: m = L%16, VGPR0 = A[m][k0 + 2*(L>=16)], VGPR1 = next K  (float2).
// B-fragment symmetric: lane L: n = L%16, VGPR0 = B[k0+2*(L>=16)][n].
// ---------------------------------------------------------------------------
__global__ void k1_gemm_relu(const float* __restrict__ x,
                             const float* __restrict__ fb,
                             float* __restrict__ p,
                             float* __restrict__ ha) {
    const int wid   = (blockIdx.x * blockDim.x + threadIdx.x) >> 5;
    const int lane  = threadIdx.x & 31;
    const int mtile = wid >> 3;          // 0..127
    const int ntile = wid & 7;           // 0..7
    const int half  = lane >> 4;         // 0|1 -> K offset 0|2, C rows 0..7|8..15
    const int l16   = lane & 15;
    const int klane = half << 1;

    const float* xrow = x + (size_t)(mtile * 16 + l16) * INF;
    const int    col  = ntile * 16 + l16;

    v8f c = {};
#pragma unroll 4
    for (int k0 = 0; k0 < INF; k0 += 4) {
        v2f a = *(const v2f*)(xrow + k0 + klane);          // contiguous K pair
        v2f b;
        b.x = fb[(k0 + klane)     * NF + col];
        b.y = fb[(k0 + klane + 1) * NF + col];
        c = __builtin_amdgcn_wmma_f32_16x16x4_f32(
                false, a, false, b, (short)0, c, false, false);
    }

    const int mbase = mtile * 16 + half * 8;
#pragma unroll
    for (int v = 0; v < 8; ++v) {
        const float av  = c[v];
        const int   idx = (mbase + v) * NF + col;
        p[idx]  = av > 0.f ? av  : 0.f;
        ha[idx] = av > 0.f ? 1.f : 0.f;
    }
}

// ---------------------------------------------------------------------------
// K2: q = relu(proto), hb = 1[proto>0]
// ---------------------------------------------------------------------------
__global__ void k2_prep(const float* __restrict__ proto,
                        float* __restrict__ q,
                        float* __restrict__ hb) {
    const int i = blockIdx.x * blockDim.x + threadIdx.x;
    if (i < OUTF * NF) {
        const float v = proto[i];
        q[i]  = v > 0.f ? v   : 0.f;
        hb[i] = v > 0.f ? 1.f : 0.f;
    }
}

// ---------------------------------------------------------------------------
// K3: S_a = p @ hb^T, S_b = ha @ q^T (M=2048, N=512, K=128) via dual WMMA.
// Prototypes stored (n,k) row-major => B fragment is a contiguous float2 load.
// Writes out = -(alpha*S_a + beta*S_b); K4 adds the inter term on top.
// ---------------------------------------------------------------------------
__global__ void k3_sab(const float* __restrict__ p,
                       const float* __restrict__ ha,
                       const float* __restrict__ q,
                       const float* __restrict__ hb,
                       const float* __restrict__ alpha_p,
                       const float* __restrict__ beta_p,
                       float* __restrict__ out) {
    const int wid   = (blockIdx.x * blockDim.x + threadIdx.x) >> 5;
    const int lane  = threadIdx.x & 31;
    const int mtile = wid >> 5;          // 0..127
    const int ntile = wid & 31;          // 0..31
    const int half  = lane >> 4;
    const int l16   = lane & 15;
    const int klane = half << 1;

    const int row = mtile * 16 + l16;
    const int col = ntile * 16 + l16;
    const float* prow  = p  + (size_t)row * NF;
    const float* harow = ha + (size_t)row * NF;
    const float* qrow  = q  + (size_t)col * NF;   // B[k][n] = q [n][k]
    const float* hbrow = hb + (size_t)col * NF;   // B[k][n] = hb[n][k]

    v8f ca = {};
    v8f cb = {};
#pragma unroll
    for (int k0 = 0; k0 < NF; k0 += 4) {
        v2f ap = *(const v2f*)(prow  + k0 + klane);
        v2f ah = *(const v2f*)(harow + k0 + klane);
        v2f bh = *(const v2f*)(hbrow + k0 + klane);
        v2f bq = *(const v2f*)(qrow  + k0 + klane);
        ca = __builtin_amdgcn_wmma_f32_16x16x4_f32(
                false, ap, false, bh, (short)0, ca, false, false);
        cb = __builtin_amdgcn_wmma_f32_16x16x4_f32(
                false, ah, false, bq, (short)0, cb, false, false);
    }

    const float alpha = alpha_p[0];
    const float beta  = beta_p[0];
    const int mbase = mtile * 16 + half * 8;
#pragma unroll
    for (int v = 0; v < 8; ++v) {
        out[(size_t)(mbase + v) * OUTF + col] = -(alpha * ca[v] + beta * cb[v]);
    }
}

// ---------------------------------------------------------------------------
// K4: inter = sum_f min(p,q); out += (theta+alpha+beta) * inter.
// One thread per (row, proto) output; float4-vectorized reduction.
// ---------------------------------------------------------------------------
__global__ void k4_inter(const float* __restrict__ p,
                         const float* __restrict__ q,
                         const float* __restrict__ theta_p,
                         const float* __restrict__ alpha_p,
                         const float* __restrict__ beta_p,
                         float* __restrict__ out) {
    const int gid = blockIdx.x * blockDim.x + threadIdx.x;   // 0 .. 2048*512-1
    const int m = gid >> 9;         // row
    const int n = gid & (OUTF - 1); // prototype
    const v4f* pr = (const v4f*)(p + (size_t)m * NF);
    const v4f* qr = (const v4f*)(q + (size_t)n * NF);

    float acc = 0.f;
#pragma unroll
    for (int f = 0; f < NF / 4; ++f) {
        const v4f pv = pr[f];
        const v4f qv = qr[f];
        acc += fminf(pv.x, qv.x) + fminf(pv.y, qv.y) +
               fminf(pv.z, qv.z) + fminf(pv.w, qv.w);
    }
    const float c = theta_p[0] + alpha_p[0] + beta_p[0];
    out[gid] = fmaf(c, acc, out[gid]);
}

// ---------------------------------------------------------------------------
// Launcher. Inputs: x, feature_bank, prototypes, theta, alpha, beta.
// Workspace layout (floats): p[2048*128] | ha[2048*128] | q[512*128] | hb[512*128]
//   = 2.5 MB total.
// ---------------------------------------------------------------------------
extern "C" void kernel_launch(void* const* d_in, const int* in_sizes, int n_in,
                              void* d_out, int out_size, void* d_ws, size_t ws_size,
                              hipStream_t stream) {
    const float* x     = (const float*)d_in[0];
    const float* fb    = (const float*)d_in[1];
    const float* proto = (const float*)d_in[2];
    const float* theta = (const float*)d_in[3];
    const float* alpha = (const float*)d_in[4];
    const float* beta  = (const float*)d_in[5];
    float* out = (float*)d_out;

    float* p  = (float*)d_ws;             // 2048*128
    float* ha = p  + ROWS * NF;           // 2048*128
    float* q  = ha + ROWS * NF;           // 512*128
    float* hb = q  + OUTF * NF;           // 512*128

    // K1: 128 M-tiles x 8 N-tiles = 1024 waves, 8 waves/block -> 128 blocks.
    k1_gemm_relu<<<128, 256, 0, stream>>>(x, fb, p, ha);

    // K2: 512*128 = 65536 elements.
    k2_prep<<<(OUTF * NF) / 256, 256, 0, stream>>>(proto, q, hb);

    // K3: 128 M-tiles x 32 N-tiles = 4096 waves -> 512 blocks.
    k3_sab<<<512, 256, 0, stream>>>(p, ha, q, hb, alpha, beta, out);

    // K4: 2048*512 = 1M outputs -> 4096 blocks.
    k4_inter<<<(ROWS * OUTF) / 256, 256, 0, stream>>>(p, q, theta, alpha, beta, out);
}